// NodeLevelGNN_14061722927248
// MI455X (gfx1250) — compile-verified
//
#include <hip/hip_runtime.h>
#include <hip/hip_bf16.h>

typedef __attribute__((ext_vector_type(2))) float v2f;
typedef __attribute__((ext_vector_type(8))) float v8f;

#define N_NODES 16384
#define N_EDGES 524288
#define DIM 32
#define MASK_WPR (N_NODES / 32)                 // 512 words per row
#define MASK_TOTAL_WORDS (N_NODES * MASK_WPR)   // 8388608 words = 32 MB

// ---------------------------------------------------------------- zero mask
__global__ void zero_mask_kernel(uint4* __restrict__ mask4, int n4) {
  int idx = blockIdx.x * blockDim.x + threadIdx.x;
  int stride = gridDim.x * blockDim.x;
  uint4 z; z.x = z.y = z.z = z.w = 0u;
  for (int i = idx; i < n4; i += stride) mask4[i] = z;
}

// ------------------------------------------------- set symmetric adjacency bits
__global__ void edge_mask_kernel(const int* __restrict__ ei,
                                 unsigned int* __restrict__ mask) {
  int e = blockIdx.x * blockDim.x + threadIdx.x;
  if (e >= N_EDGES) return;
  int s = ei[e];
  int d = ei[N_EDGES + e];
  atomicOr(&mask[(size_t)s * MASK_WPR + (d >> 5)], 1u << (d & 31));
  atomicOr(&mask[(size_t)d * MASK_WPR + (s >> 5)], 1u << (s & 31));
}

// ---------------------------------------- sparse exact aggregation, 1 wave/row
// lane l accumulates messages[row][l]  (DIM == 32 == wave width)
__global__ void aggregate_kernel(const unsigned int* __restrict__ mask,
                                 const float* __restrict__ x,
                                 float* __restrict__ msg) {
  int wave = (blockIdx.x * blockDim.x + threadIdx.x) >> 5;
  int lane = threadIdx.x & 31;
  if (wave >= N_NODES) return;
  const unsigned int* row = mask + (size_t)wave * MASK_WPR;
  float acc = 0.0f;
#pragma unroll 1
  for (int it = 0; it < MASK_WPR / 32; ++it) {
    unsigned int w = row[it * 32 + lane];        // coalesced 128B per iter
    unsigned long long nz = __ballot(w != 0u);   // wave32: low 32 bits
    while (nz) {
      int sl = __ffsll(nz) - 1;
      nz &= nz - 1;
      unsigned int ww = __shfl(w, sl);           // uniform word
      int jbase = (it * 32 + sl) * 32;
      while (ww) {
        int b = __ffs((int)ww) - 1;
        ww &= ww - 1;
        acc += x[(size_t)(jbase + b) * DIM + lane];  // coalesced 128B gather
      }
    }
  }
  msg[(size_t)wave * DIM + lane] = acc;
}

// -------- fused exact f32 WMMA: out = x@Wupd^T + msg@Wagg^T + (b_upd+b_agg)
// one wave computes a 16-row x 32-col output tile via V_WMMA_F32_16X16X4_F32
__global__ void gemm_out_kernel(const float* __restrict__ x,
                                const float* __restrict__ msg,
                                const float* __restrict__ Wagg,
                                const float* __restrict__ bagg,
                                const float* __restrict__ Wupd,
                                const float* __restrict__ bupd,
                                float* __restrict__ out) {
  int wave = (blockIdx.x * blockDim.x + threadIdx.x) >> 5;  // 0..1023 row tiles
  int lane = threadIdx.x & 31;
  if (wave >= N_NODES / 16) return;
  int row0  = wave * 16;
  int lhalf = lane >> 4;   // 0: K/M low half, 1: high half
  int l15   = lane & 15;

#pragma unroll
  for (int c0 = 0; c0 < DIM; c0 += 16) {
    // C/D element (m, n) lives in VGPR v = m % 8, lane = (m/8)*16 + n
    float bias = bupd[c0 + l15] + bagg[c0 + l15];
    v8f acc;
#pragma unroll
    for (int v = 0; v < 8; ++v) acc[v] = bias;

    // x @ Wupd^T  : A[m][k] = x[row0+m][k],  B[k][n] = Wupd[c0+n][k]
#pragma unroll
    for (int k0 = 0; k0 < DIM; k0 += 4) {
      // A 16x4 f32: lanes 0-15 -> (K=k0, k0+1), lanes 16-31 -> (K=k0+2, k0+3)
      v2f a = *(const v2f*)&x[(size_t)(row0 + l15) * DIM + k0 + lhalf * 2];
      v2f b = *(const v2f*)&Wupd[(size_t)(c0 + l15) * DIM + k0 + lhalf * 2];
      acc = __builtin_amdgcn_wmma_f32_16x16x4_f32(false, a, false, b,
                                                  (short)0, acc, false, false);
    }
    // msg @ Wagg^T
#pragma unroll
    for (int k0 = 0; k0 < DIM; k0 += 4) {
      v2f a = *(const v2f*)&msg[(size_t)(row0 + l15) * DIM + k0 + lhalf * 2];
      v2f b = *(const v2f*)&Wagg[(size_t)(c0 + l15) * DIM + k0 + lhalf * 2];
      acc = __builtin_amdgcn_wmma_f32_16x16x4_f32(false, a, false, b,
                                                  (short)0, acc, false, false);
    }
#pragma unroll
    for (int v = 0; v < 8; ++v)
      out[(size_t)(row0 + lhalf * 8 + v) * DIM + c0 + l15] = acc[v];
  }
}

// ---------------------------------------------------------------- launcher
extern "C" void kernel_launch(void* const* d_in, const int* in_sizes, int n_in,
                              void* d_out, int out_size, void* d_ws, size_t ws_size,
                              hipStream_t stream) {
  const float* x    = (const float*)d_in[0];
  const int*   ei   = (const int*)d_in[1];
  const float* Wagg = (const float*)d_in[2];
  const float* bagg = (const float*)d_in[3];
  const float* Wupd = (const float*)d_in[4];
  const float* bupd = (const float*)d_in[5];
  float* out = (float*)d_out;

  unsigned int* mask = (unsigned int*)d_ws;                       // 32 MB
  float* msg = (float*)((char*)d_ws + (size_t)MASK_TOTAL_WORDS * 4); // 2 MB

  zero_mask_kernel<<<2048, 256, 0, stream>>>((uint4*)mask, MASK_TOTAL_WORDS / 4);

  edge_mask_kernel<<<(N_EDGES + 255) / 256, 256, 0, stream>>>(ei, mask);

  // one wave per node row: N_NODES waves = N_NODES*32 threads
  aggregate_kernel<<<(N_NODES * 32 + 255) / 256, 256, 0, stream>>>(mask, x, msg);

  // one wave per 16-row output tile: 1024 waves
  gemm_out_kernel<<<(N_NODES / 16 * 32 + 255) / 256, 256, 0, stream>>>(
      x, msg, Wagg, bagg, Wupd, bupd, out);
}